// RegionSparsityGate_79474074845628
// MI455X (gfx1250) — compile-verified
//
#include <hip/hip_runtime.h>
#include <math.h>

typedef __attribute__((ext_vector_type(2))) float        v2f;
typedef __attribute__((ext_vector_type(8))) float        v8f;
typedef __attribute__((ext_vector_type(4))) unsigned int v4u;
typedef __attribute__((ext_vector_type(8))) int          v8i;
typedef __attribute__((ext_vector_type(4))) int          v4i;

#define R_N   1024
#define B_N   16
#define D_N   1024
#define CHUNK 256
#define K_ACT 20
#define NEG_INF (-3.402823466e38f)

// ---------------------------------------------------------------------------
// Kernel A: adj[b,r] = s - theta - refr - 0.5*(0.9*fb + 0.1*||msg||)
//   s via V_WMMA_F32_16X16X4_F32, H tile staged to LDS by the Tensor Data Mover
//   One block (4 waves) per region; wave w covers K slice [w*64, w*64+64) of
//   each 256-wide chunk.
// ---------------------------------------------------------------------------
__global__ __launch_bounds__(128) void rsg_adj_kernel(
    const float* __restrict__ H, const float* __restrict__ msg,
    const float* __restrict__ w, const float* __restrict__ theta,
    const float* __restrict__ refr, const float* __restrict__ fb,
    float* __restrict__ out_adj)
{
  __shared__ float Htile[B_N * CHUNK];   // 16 rows x 256 K, 16 KB
  __shared__ float s_part[4 * 16];
  __shared__ float m_part[128];

  const int r    = blockIdx.x;
  const int tid  = threadIdx.x;
  const int lane = tid & 31;
  const int wid  = tid >> 5;
  const int hi   = lane >> 4;     // half-wave: K pair select
  const int M    = lane & 15;     // A-matrix row = batch index

  const float* Hr = H + (size_t)r * B_N * D_N;

  v8f c = {};   // 16x16 f32 accumulator: s[b] replicated across columns

  for (int ch = 0; ch < 4; ++ch) {
    __syncthreads();   // all waves done reading previous chunk
    if (wid == 0) {
#if __has_builtin(__builtin_amdgcn_tensor_load_to_lds)
      // --- TDM descriptor: 2D tile 256(K) x 16(B) from tensor stride 1024 ---
      unsigned long long ga = (unsigned long long)(const void*)(Hr + ch * CHUNK);
      unsigned lds_addr = (unsigned)(unsigned long long)(const void*)&Htile[0];
      v4u g0;
      g0.x = 1u;                                   // count=1, user descriptor
      g0.y = lds_addr;                             // LDS byte address
      g0.z = (unsigned)(ga & 0xFFFFFFFFu);         // global_addr[31:0]
      g0.w = (unsigned)((ga >> 32) & 0x1FFFFFFu)   // global_addr[56:32]
           | (2u << 30);                           // type = 2 ("image")
      v8i g1;
      g1[0] = 0x00020000;                          // data_size = 4 bytes
      g1[1] = (D_N & 0xFFFF) << 16;                // tensor_dim0[15:0]
      g1[2] = (D_N >> 16) | (B_N << 16);           // tensor_dim0[31:16] | tensor_dim1[15:0]
      g1[3] = (CHUNK << 16);                       // tile_dim0 = 256
      g1[4] = B_N;                                 // tile_dim1 = 16
      g1[5] = D_N;                                 // tensor_dim0_stride[31:0] = 1024
      g1[6] = 0;                                   // stride hi | tensor_dim1_stride lo (2D: unused)
      g1[7] = 0;
      v4i gz4 = {0, 0, 0, 0};
      v8i gz8 = {0, 0, 0, 0, 0, 0, 0, 0};
      // amdgpu-toolchain (clang-23) arity: 6 args, extra int32x8 group (zeros)
      __builtin_amdgcn_tensor_load_to_lds(g0, g1, gz4, gz4, gz8, 0);
      __builtin_amdgcn_s_wait_tensorcnt(0);
#else
      for (int i = lane; i < B_N * CHUNK; i += 32) {
        int bb = i / CHUNK, dd = i % CHUNK;
        Htile[i] = Hr[bb * D_N + ch * CHUNK + dd];
      }
#endif
    }
    __syncthreads();   // tile visible to all 4 waves

    const int kbase = wid * 64;
    #pragma unroll 4
    for (int k0 = 0; k0 < 64; k0 += 4) {
      // 32-bit A 16x4 layout: lanes 0-15 -> K{0,1}, lanes 16-31 -> K{2,3}
      const int k = kbase + k0 + 2 * hi;
      v2f a, bm;
      a.x  = Htile[M * CHUNK + k];
      a.y  = Htile[M * CHUNK + k + 1];
      bm.x = w[ch * CHUNK + k];       // B row K, replicated over all N columns
      bm.y = w[ch * CHUNK + k + 1];
      c = __builtin_amdgcn_wmma_f32_16x16x4_f32(
              false, a, false, bm, (short)0, c, false, false);
    }
  }

  // Extract s partials: column N==0 (lanes 0 and 16); VGPR v holds row v+8*hi
  if ((lane & 15) == 0) {
    #pragma unroll
    for (int v = 0; v < 8; ++v) s_part[wid * 16 + hi * 8 + v] = c[v];
  }

  // ||msg[r,b,:]||^2 : 8 threads per batch row, float4 streaming
  {
    const float4* Mr4 = (const float4*)(msg + (size_t)r * B_N * D_N);
    const int b = tid >> 3;        // 0..15
    const int g = tid & 7;         // 0..7
    float acc = 0.f;
    #pragma unroll 4
    for (int j = 0; j < 32; ++j) {
      float4 x = Mr4[b * 256 + g + j * 8];
      acc += x.x * x.x + x.y * x.y + x.z * x.z + x.w * x.w;
    }
    m_part[tid] = acc;
  }
  __syncthreads();

  if (tid < 16) {
    float s = 0.f;
    #pragma unroll
    for (int wv = 0; wv < 4; ++wv) s += s_part[wv * 16 + tid];
    float msq = 0.f;
    #pragma unroll
    for (int g = 0; g < 8; ++g) msq += m_part[tid * 8 + g];
    const float fbn  = 0.9f * fb[tid * R_N + r] + 0.1f * sqrtf(msq);
    const float adjv = s - theta[r] - refr[tid * R_N + r] - 0.5f * fbn;
    out_adj[tid * R_N + r] = adjv;
  }
}

// ---------------------------------------------------------------------------
// Kernel B: greedy hex NMS. Sorted-order greedy == repeated argmax over
// non-suppressed candidates. One block per batch.
// ---------------------------------------------------------------------------
__global__ __launch_bounds__(256) void rsg_nms_kernel(
    const float* __restrict__ adj, const int* __restrict__ nbrs,
    float* __restrict__ out_hard)
{
  __shared__ float score[R_N];
  __shared__ float hardm[R_N];
  __shared__ float rv[256];
  __shared__ int   ri[256];

  const int b = blockIdx.x;
  const int t = threadIdx.x;

  for (int i = t; i < R_N; i += 256) {
    score[i] = adj[b * R_N + i];
    hardm[i] = 0.f;
  }
  __syncthreads();

  for (int it = 0; it < K_ACT; ++it) {
    float best = NEG_INF; int bi = -1;
    for (int i = t; i < R_N; i += 256) {
      float v = score[i];
      if (v > best) { best = v; bi = i; }
    }
    rv[t] = best; ri[t] = bi;
    __syncthreads();
    for (int off = 128; off > 0; off >>= 1) {
      if (t < off && rv[t + off] > rv[t]) { rv[t] = rv[t + off]; ri[t] = ri[t + off]; }
      __syncthreads();
    }
    if (t == 0) {
      const int idx = ri[0];
      if (idx >= 0 && rv[0] > -1e30f) {
        hardm[idx] = 1.f;
        score[idx] = NEG_INF;
        #pragma unroll
        for (int n = 0; n < 6; ++n) score[nbrs[idx * 6 + n]] = NEG_INF;
      }
    }
    __syncthreads();
  }

  for (int i = t; i < R_N; i += 256)
    out_hard[b * R_N + i] = hardm[i];
}

// ---------------------------------------------------------------------------
// Kernel C: Hs[r,b,d] = H[r,b,d] * hard[b,r]   (ste == hard numerically)
// Pure streaming, float4, one block per region.
// ---------------------------------------------------------------------------
__global__ __launch_bounds__(256) void rsg_gate_kernel(
    const float* __restrict__ H, const float* __restrict__ hard,
    float* __restrict__ outHs)
{
  __shared__ float hb[B_N];
  const int r = blockIdx.x;
  const int t = threadIdx.x;
  if (t < B_N) hb[t] = hard[t * R_N + r];
  __syncthreads();

  const float4* H4 = (const float4*)(H + (size_t)r * B_N * D_N);
  float4*       O4 = (float4*)(outHs + (size_t)r * B_N * D_N);

  #pragma unroll
  for (int j = 0; j < 16; ++j) {
    const int f = j * 256 + t;          // 4096 float4 per region
    if (j + 4 < 16) __builtin_prefetch(&H4[f + 4 * 256], 0, 0);
    const float  g = hb[f >> 8];        // 256 float4 per batch row
    const float4 x = H4[f];
    float4 y; y.x = x.x * g; y.y = x.y * g; y.z = x.z * g; y.w = x.w * g;
    O4[f] = y;
  }
}

// ---------------------------------------------------------------------------
extern "C" void kernel_launch(void* const* d_in, const int* in_sizes, int n_in,
                              void* d_out, int out_size, void* d_ws, size_t ws_size,
                              hipStream_t stream) {
  const float* H     = (const float*)d_in[0];
  const float* msg   = (const float*)d_in[1];
  const float* w     = (const float*)d_in[2];
  const float* theta = (const float*)d_in[3];
  const float* refr  = (const float*)d_in[4];
  const float* fb    = (const float*)d_in[5];
  const int*   nbrs  = (const int*)d_in[6];

  float* out      = (float*)d_out;
  float* out_Hs   = out;                                   // [R,B,D]
  float* out_hard = out + (size_t)R_N * B_N * D_N;         // [B,R]
  float* out_adj  = out_hard + (size_t)B_N * R_N;          // [B,R]

  rsg_adj_kernel <<<R_N, 128, 0, stream>>>(H, msg, w, theta, refr, fb, out_adj);
  rsg_nms_kernel <<<B_N, 256, 0, stream>>>(out_adj, nbrs, out_hard);
  rsg_gate_kernel<<<R_N, 256, 0, stream>>>(H, out_hard, out_Hs);
}